// ObjectRelationNet_46256797778375
// MI455X (gfx1250) — compile-verified
//
#include <hip/hip_runtime.h>

// Fused ObjectRelationNet: out = (relu(relu(pairs@W1+b1)@W2+b2))@W3+b3
// pairs[p] = concat(obj[i], obj[j]) for p-th (i<j) pair, gathered on the fly.
// fp32 end-to-end via V_WMMA_F32_16X16X4_F32 (exact fp32 semantics).
// Each wave computes BOTH 16-row M-tiles for its N-tiles, so every weight
// (B) fragment load feeds two WMMAs: 8 VMEM : 6 WMMA per K-step.

#define NOBJ   256
#define HID    768
#define NPAIRS 32640   // 256*255/2
#define TM     32      // pair rows per workgroup
#define H1S    772     // padded LDS row stride (floats) for h1  (768+4)
#define H2S    388     // padded LDS row stride (floats) for h2  (384+4)

typedef float v2f __attribute__((ext_vector_type(2)));
typedef float v8f __attribute__((ext_vector_type(8)));

__device__ __forceinline__ v8f wmma4(v8f c, v2f a, v2f b) {
  // D = A(16x4 f32) x B(4x16 f32) + C(16x16 f32)
  return __builtin_amdgcn_wmma_f32_16x16x4_f32(false, a, false, b, (short)0, c,
                                               false, false);
}

__device__ __forceinline__ v8f splat8(float x) {
  v8f r = {x, x, x, x, x, x, x, x};
  return r;
}

extern "C" __global__ __launch_bounds__(256)
void relnet_fused(const float* __restrict__ obj,
                  const float* __restrict__ W1, const float* __restrict__ B1,
                  const float* __restrict__ W2, const float* __restrict__ B2,
                  const float* __restrict__ W3, const float* __restrict__ B3,
                  float* __restrict__ out) {
  __shared__ float h1[TM * H1S];   //  98,816 B
  __shared__ float h2[TM * H2S];   //  49,664 B
  __shared__ int   s_i[TM];
  __shared__ int   s_j[TM];

  const int tid    = threadIdx.x;
  const int wave   = tid >> 5;     // 0..7
  const int lane   = tid & 31;
  const int lane16 = lane & 15;
  const int lg     = lane >> 4;    // fragment lane-group (0/1)
  const int ko     = 2 * lg;       // K offset inside a 16x4 fragment

  // Decode triu(i<j) pair indices for this workgroup's 32 rows.
  if (tid < TM) {
    int p = blockIdx.x * TM + tid;
    int rem = p, i = 0;
    while (rem >= (NOBJ - 1 - i)) { rem -= (NOBJ - 1 - i); ++i; }
    s_i[tid] = i;
    s_j[tid] = i + 1 + rem;
  }
  __syncthreads();

  // A-operand source rows for this lane: M-tile0 row = lane16,
  // M-tile1 row = 16+lane16.
  const float* pi0 = obj + s_i[lane16] * HID;
  const float* pj0 = obj + s_j[lane16] * HID;
  const float* pi1 = obj + s_i[16 + lane16] * HID;
  const float* pj1 = obj + s_j[16 + lane16] * HID;

  // ---------------- Layer 1: h1 = relu(pairs @ W1 + b1), 32x768 ----------
  for (int pass = 0; pass < 2; ++pass) {
    const int nb = (pass * 8 + wave) * 48;         // 3 N-tiles start here
    const float bv0 = B1[nb +  0 + lane16];
    const float bv1 = B1[nb + 16 + lane16];
    const float bv2 = B1[nb + 32 + lane16];
    v8f a00 = splat8(bv0), a01 = splat8(bv1), a02 = splat8(bv2);  // M-tile 0
    v8f a10 = splat8(bv0), a11 = splat8(bv1), a12 = splat8(bv2);  // M-tile 1

    // K in [0,768): A comes from object i
#pragma unroll 4
    for (int kb = 0; kb < HID; kb += 4) {
      v2f av0 = *(const v2f*)(pi0 + kb + ko);
      v2f av1 = *(const v2f*)(pi1 + kb + ko);
      const float* wp = W1 + (kb + ko) * HID + nb + lane16;
      v2f bf0 = {wp[0],  wp[HID]};
      v2f bf1 = {wp[16], wp[HID + 16]};
      v2f bf2 = {wp[32], wp[HID + 32]};
      a00 = wmma4(a00, av0, bf0);
      a10 = wmma4(a10, av1, bf0);
      a01 = wmma4(a01, av0, bf1);
      a11 = wmma4(a11, av1, bf1);
      a02 = wmma4(a02, av0, bf2);
      a12 = wmma4(a12, av1, bf2);
    }
    // K in [768,1536): A comes from object j
#pragma unroll 4
    for (int kb = 0; kb < HID; kb += 4) {
      v2f av0 = *(const v2f*)(pj0 + kb + ko);
      v2f av1 = *(const v2f*)(pj1 + kb + ko);
      const float* wp = W1 + (HID + kb + ko) * HID + nb + lane16;
      v2f bf0 = {wp[0],  wp[HID]};
      v2f bf1 = {wp[16], wp[HID + 16]};
      v2f bf2 = {wp[32], wp[HID + 32]};
      a00 = wmma4(a00, av0, bf0);
      a10 = wmma4(a10, av1, bf0);
      a01 = wmma4(a01, av0, bf1);
      a11 = wmma4(a11, av1, bf1);
      a02 = wmma4(a02, av0, bf2);
      a12 = wmma4(a12, av1, bf2);
    }
#pragma unroll
    for (int v = 0; v < 8; ++v) {                  // relu + spill to LDS
      const int r0 = lg * 8 + v;
      const int r1 = 16 + lg * 8 + v;
      h1[r0 * H1S + nb +  0 + lane16] = fmaxf(a00[v], 0.f);
      h1[r0 * H1S + nb + 16 + lane16] = fmaxf(a01[v], 0.f);
      h1[r0 * H1S + nb + 32 + lane16] = fmaxf(a02[v], 0.f);
      h1[r1 * H1S + nb +  0 + lane16] = fmaxf(a10[v], 0.f);
      h1[r1 * H1S + nb + 16 + lane16] = fmaxf(a11[v], 0.f);
      h1[r1 * H1S + nb + 32 + lane16] = fmaxf(a12[v], 0.f);
    }
  }
  __syncthreads();

  // ---------------- Layer 2: h2 = relu(h1 @ W2 + b2), 32x384 -------------
  {
    const int nb = wave * 48;                      // 3 N-tiles per wave
    const float bv0 = B2[nb +  0 + lane16];
    const float bv1 = B2[nb + 16 + lane16];
    const float bv2 = B2[nb + 32 + lane16];
    v8f a00 = splat8(bv0), a01 = splat8(bv1), a02 = splat8(bv2);
    v8f a10 = splat8(bv0), a11 = splat8(bv1), a12 = splat8(bv2);
#pragma unroll 4
    for (int kb = 0; kb < HID; kb += 4) {
      v2f av0 = *(const v2f*)&h1[lane16 * H1S + kb + ko];
      v2f av1 = *(const v2f*)&h1[(16 + lane16) * H1S + kb + ko];
      const float* wp = W2 + (kb + ko) * 384 + nb + lane16;
      v2f bf0 = {wp[0],  wp[384]};
      v2f bf1 = {wp[16], wp[384 + 16]};
      v2f bf2 = {wp[32], wp[384 + 32]};
      a00 = wmma4(a00, av0, bf0);
      a10 = wmma4(a10, av1, bf0);
      a01 = wmma4(a01, av0, bf1);
      a11 = wmma4(a11, av1, bf1);
      a02 = wmma4(a02, av0, bf2);
      a12 = wmma4(a12, av1, bf2);
    }
#pragma unroll
    for (int v = 0; v < 8; ++v) {
      const int r0 = lg * 8 + v;
      const int r1 = 16 + lg * 8 + v;
      h2[r0 * H2S + nb +  0 + lane16] = fmaxf(a00[v], 0.f);
      h2[r0 * H2S + nb + 16 + lane16] = fmaxf(a01[v], 0.f);
      h2[r0 * H2S + nb + 32 + lane16] = fmaxf(a02[v], 0.f);
      h2[r1 * H2S + nb +  0 + lane16] = fmaxf(a10[v], 0.f);
      h2[r1 * H2S + nb + 16 + lane16] = fmaxf(a11[v], 0.f);
      h2[r1 * H2S + nb + 32 + lane16] = fmaxf(a12[v], 0.f);
    }
  }
  __syncthreads();

  // ---------------- Layer 3: out = h2 @ W3 + b3, 32x128 ------------------
  {
    const int nb = wave * 16;                      // 1 N-tile per wave
    const float bv = B3[nb + lane16];
    v8f a0 = splat8(bv);                           // M-tile 0
    v8f a1 = splat8(bv);                           // M-tile 1
#pragma unroll 4
    for (int kb = 0; kb < 384; kb += 4) {
      v2f av0 = *(const v2f*)&h2[lane16 * H2S + kb + ko];
      v2f av1 = *(const v2f*)&h2[(16 + lane16) * H2S + kb + ko];
      const float* wp = W3 + (kb + ko) * 128 + nb + lane16;
      v2f bf = {wp[0], wp[128]};
      a0 = wmma4(a0, av0, bf);
      a1 = wmma4(a1, av1, bf);
    }
    const int pbase = blockIdx.x * TM;
#pragma unroll
    for (int v = 0; v < 8; ++v) {
      const int r0 = lg * 8 + v;
      const int r1 = 16 + lg * 8 + v;
      out[(pbase + r0) * 128 + nb + lane16] = a0[v];
      out[(pbase + r1) * 128 + nb + lane16] = a1[v];
    }
  }
}

extern "C" void kernel_launch(void* const* d_in, const int* in_sizes, int n_in,
                              void* d_out, int out_size, void* d_ws,
                              size_t ws_size, hipStream_t stream) {
  (void)in_sizes; (void)n_in; (void)d_ws; (void)ws_size; (void)out_size;
  const float* obj = (const float*)d_in[0];
  const float* W1  = (const float*)d_in[1];
  const float* B1  = (const float*)d_in[2];
  const float* W2  = (const float*)d_in[3];
  const float* B2  = (const float*)d_in[4];
  const float* W3  = (const float*)d_in[5];
  const float* B3  = (const float*)d_in[6];
  float* out = (float*)d_out;

  dim3 grid(NPAIRS / TM);   // 1020 workgroups
  dim3 block(256);          // 8 wave32s
  hipLaunchKernelGGL(relnet_fused, grid, block, 0, stream,
                     obj, W1, B1, W2, B2, W3, B3, out);
}